// rGCN_45646912422571
// MI455X (gfx1250) — compile-verified
//
#include <hip/hip_runtime.h>
#include <hip/hip_bf16.h>

typedef float v2f __attribute__((ext_vector_type(2)));
typedef float v8f __attribute__((ext_vector_type(8)));
typedef int   v4i __attribute__((vector_size(4 * sizeof(int))));

#define NROW 8192
#define RREL 4
#define NHID 16
#define NCLS 8
#define INV_NEIGH 0.1f

// workspace layout (float offsets)
#define WS_XW1 0                               // [R][N][16]
#define WS_XW2 (WS_XW1 + RREL * NROW * NHID)   // [R][N][16] (cols 8..15 zero-padded)
#define WS_AGG (WS_XW2 + RREL * NROW * NHID)   // [N][16] accumulator (reused per layer)
#define WS_H   (WS_AGG + NROW * NHID)          // [N][16]

// CDNA5 async global->LDS path (ASYNCcnt-tracked), guarded so a missing
// builtin falls back to the proven direct-load path.
#if defined(__gfx1250__) && \
    __has_builtin(__builtin_amdgcn_global_load_async_to_lds_b128) && \
    __has_builtin(__builtin_amdgcn_s_wait_asynccnt)
#define HAVE_ASYNC_LDS 1
#else
#define HAVE_ASYNC_LDS 0
#endif

#define TM      64                // m-extent of one staged A tile
#define ASTRIDE 68                // padded LDS row stride (floats): 272 B = 17*16 B
                                  // (keeps b128 dests 16B-aligned) and 68 mod 64 = 4
                                  // -> conflict-free bank mapping for the WMMA reads
#define ABUF    (16 * ASTRIDE)    // floats per LDS buffer (4352 B, 16B multiple)

// ---------------------------------------------------------------------------
// Kernel 1: xw1[r][m][h] = sum_f x[m][f] * W1[r][f][h]
// One wave per (16-row m-tile, relation). WMMA f32 16x16x4 chain over f.
// A layout (16x4 f32): lane L<16 -> M=L, K={0,1}; lane L>=16 -> M=L-16, K={2,3}
// B layout (4x16 f32): lane L<16 -> N=L, K={0,1}; lane L>=16 -> N=L-16, K={2,3}
// C/D layout: VGPR j, lane L -> (M = j + 8*(L>>4), N = L&15)
// ---------------------------------------------------------------------------
__global__ __launch_bounds__(32) void k_xw1(const float* __restrict__ x,
                                            const float* __restrict__ W1,
                                            float* __restrict__ xw1) {
    const int mtile = blockIdx.x;        // 0..511
    const int r     = blockIdx.y;        // 0..3
    const int lane  = threadIdx.x & 31;
    const int col   = lane & 15;         // M for A, N for B/D
    const int khi   = lane >> 4;         // 0/1 -> K base 0 or 2
    const int mbase = mtile * 16;

    const float* xrow  = x + (size_t)(mbase + col) * NROW + 2 * khi;
    const float* wbase = W1 + ((size_t)r * NROW + 2 * khi) * NHID + col;

    v8f acc = {};
#pragma unroll 4
    for (int f = 0; f < NROW; f += 4) {
        float2 a2 = *(const float2*)(xrow + f);
        v2f a = {a2.x, a2.y};
        v2f b = {wbase[(size_t)f * NHID], wbase[(size_t)f * NHID + NHID]};
        acc = __builtin_amdgcn_wmma_f32_16x16x4_f32(false, a, false, b,
                                                    (short)0, acc, false, false);
    }

    float* o = xw1 + ((size_t)r * NROW + mbase + 8 * khi) * NHID + col;
#pragma unroll
    for (int j = 0; j < 8; ++j) o[(size_t)j * NHID] = acc[j];
}

// ---------------------------------------------------------------------------
// Kernel 2/4 workhorse: agg[n][h] += sum_r sum_{m in chunk} adjs[r][n][m]*v[r][m][h]
// One wave per (16-row n-tile, m-chunk). A (the 1 GB adjs stream) is staged
// through LDS with double-buffered async b128 copies (512 B coalesced per
// instruction, ASYNCcnt-pipelined); WMMA consumes A from LDS (ds_load_b64,
// bank-conflict-free) while the next tile streams in. Partial tiles are
// combined across m-chunks with global f32 atomics.
// ---------------------------------------------------------------------------
__global__ __launch_bounds__(32) void k_agg(const float* __restrict__ adjs,
                                            const float* __restrict__ v,   // [R][N][16]
                                            float* __restrict__ agg,       // [N][16]
                                            int mchunk) {
    const int ntile = blockIdx.x;        // 0..511
    const int chunk = blockIdx.y;
    const int lane  = threadIdx.x & 31;
    const int col   = lane & 15;
    const int khi   = lane >> 4;
    const int nbase = ntile * 16;
    const int m0    = chunk * mchunk;

    v8f acc = {};

#if HAVE_ASYNC_LDS
    __shared__ __align__(16) float abuf[2 * ABUF];

    // Stage one 16-row x TM-col f32 tile: 8 async b128 ops, each lane copies
    // one 16 B chunk; rows are contiguous 256 B global runs (fully coalesced).
    auto prefetch = [&](const float* gbase, float* lbase) {
#pragma unroll
        for (int i = 0; i < 8; ++i) {
            const int c    = lane + 32 * i;     // chunk 0..255
            const int row  = c >> 4;            // 0..15
            const int coff = (c & 15) * 4;      // float offset in row, 0..60
            __builtin_amdgcn_global_load_async_to_lds_b128(
                (__attribute__((address_space(1))) v4i*)(gbase + (size_t)row * NROW + coff),
                (__attribute__((address_space(3))) v4i*)(lbase + row * ASTRIDE + coff),
                0, 0);
        }
    };

    const int nt = mchunk / TM;                 // tiles per chunk (8 for mchunk=512)
    for (int r = 0; r < RREL; ++r) {
        const float* aBase = adjs + ((size_t)r * NROW + nbase) * NROW + m0;
        prefetch(aBase, abuf);
        for (int t = 0; t < nt; ++t) {
            if (t + 1 < nt) {
                prefetch(aBase + (t + 1) * TM, abuf + ((t + 1) & 1) * ABUF);
                __builtin_amdgcn_s_wait_asynccnt(8);   // tile t landed (in-order)
            } else {
                __builtin_amdgcn_s_wait_asynccnt(0);
            }
            const float* al = abuf + (t & 1) * ABUF + col * ASTRIDE + 2 * khi;
            const float* bb = v + ((size_t)r * NROW + m0 + t * TM + 2 * khi) * NHID + col;
#pragma unroll
            for (int mm = 0; mm < TM; mm += 4) {
                float2 a2 = *(const float2*)(al + mm);           // ds_load_b64
                v2f a = {a2.x, a2.y};
                v2f b = {bb[(size_t)mm * NHID], bb[(size_t)mm * NHID + NHID]};
                acc = __builtin_amdgcn_wmma_f32_16x16x4_f32(false, a, false, b,
                                                            (short)0, acc, false, false);
            }
        }
    }
#else
    for (int r = 0; r < RREL; ++r) {
        const float* arow = adjs + ((size_t)r * NROW + nbase + col) * NROW + m0 + 2 * khi;
        const float* vb   = v + ((size_t)r * NROW + m0 + 2 * khi) * NHID + col;
#pragma unroll 4
        for (int m = 0; m < mchunk; m += 4) {
            float2 a2 = *(const float2*)(arow + m);
            v2f a = {a2.x, a2.y};
            v2f b = {vb[(size_t)m * NHID], vb[(size_t)m * NHID + NHID]};
            acc = __builtin_amdgcn_wmma_f32_16x16x4_f32(false, a, false, b,
                                                        (short)0, acc, false, false);
        }
    }
#endif

    float* o = agg + (size_t)(nbase + 8 * khi) * NHID + col;
#pragma unroll
    for (int j = 0; j < 8; ++j) atomicAdd(o + (size_t)j * NHID, acc[j]);
}

// h = relu(agg / num_neighbors + b1)
__global__ void k_relu_bias(const float* __restrict__ agg, const float* __restrict__ b1,
                            float* __restrict__ h) {
    int i = blockIdx.x * blockDim.x + threadIdx.x;
    if (i < NROW * NHID) {
        float val = agg[i] * INV_NEIGH + b1[i & (NHID - 1)];
        h[i] = val > 0.f ? val : 0.f;
    }
}

// xw2[r][m][c] = sum_h h[m][h] * W2[r][h][c]; pad cols 8..15 with zeros
__global__ void k_xw2(const float* __restrict__ h, const float* __restrict__ W2,
                      float* __restrict__ xw2) {
    int i = blockIdx.x * blockDim.x + threadIdx.x;   // over R*N
    if (i >= RREL * NROW) return;
    int r = i / NROW, m = i - r * NROW;
    const float* hr = h + (size_t)m * NHID;
    const float* w  = W2 + r * NHID * NCLS;
    float outv[NCLS];
#pragma unroll
    for (int c = 0; c < NCLS; ++c) outv[c] = 0.f;
#pragma unroll
    for (int k = 0; k < NHID; ++k) {
        float hv = hr[k];
#pragma unroll
        for (int c = 0; c < NCLS; ++c) outv[c] = fmaf(hv, w[k * NCLS + c], outv[c]);
    }
    float* o = xw2 + ((size_t)r * NROW + m) * NHID;
#pragma unroll
    for (int c = 0; c < NCLS; ++c) o[c] = outv[c];
#pragma unroll
    for (int c = NCLS; c < NHID; ++c) o[c] = 0.f;
}

// out[n][c] = log_softmax(agg[n][0..7]/num_neighbors + b2)
__global__ void k_logsoftmax(const float* __restrict__ agg, const float* __restrict__ b2,
                             float* __restrict__ out) {
    int n = blockIdx.x * blockDim.x + threadIdx.x;
    if (n >= NROW) return;
    float y[NCLS];
    float mx = -3.0e38f;
#pragma unroll
    for (int c = 0; c < NCLS; ++c) {
        y[c] = agg[(size_t)n * NHID + c] * INV_NEIGH + b2[c];
        mx = fmaxf(mx, y[c]);
    }
    float s = 0.f;
#pragma unroll
    for (int c = 0; c < NCLS; ++c) s += __expf(y[c] - mx);
    float lse = mx + __logf(s);
#pragma unroll
    for (int c = 0; c < NCLS; ++c) out[(size_t)n * NCLS + c] = y[c] - lse;
}

extern "C" void kernel_launch(void* const* d_in, const int* in_sizes, int n_in,
                              void* d_out, int out_size, void* d_ws, size_t ws_size,
                              hipStream_t stream) {
    const float* x    = (const float*)d_in[0];
    const float* adjs = (const float*)d_in[1];
    const float* W1   = (const float*)d_in[2];
    const float* b1   = (const float*)d_in[3];
    const float* W2   = (const float*)d_in[4];
    const float* b2   = (const float*)d_in[5];
    float* out = (float*)d_out;
    float* ws  = (float*)d_ws;

    float* xw1 = ws + WS_XW1;
    float* xw2 = ws + WS_XW2;
    float* agg = ws + WS_AGG;
    float* h   = ws + WS_H;

    const int MCHUNK_BLKS = 16;              // 8192 waves in k_agg for occupancy
    const int mchunk = NROW / MCHUNK_BLKS;   // 512 (multiple of TM)

    // ---- layer 1 ----
    k_xw1<<<dim3(NROW / 16, RREL), 32, 0, stream>>>(x, W1, xw1);
    (void)hipMemsetAsync(agg, 0, (size_t)NROW * NHID * sizeof(float), stream);
    k_agg<<<dim3(NROW / 16, MCHUNK_BLKS), 32, 0, stream>>>(adjs, xw1, agg, mchunk);
    k_relu_bias<<<(NROW * NHID + 255) / 256, 256, 0, stream>>>(agg, b1, h);

    // ---- layer 2 ----
    k_xw2<<<(RREL * NROW + 255) / 256, 256, 0, stream>>>(h, W2, xw2);
    (void)hipMemsetAsync(agg, 0, (size_t)NROW * NHID * sizeof(float), stream);
    k_agg<<<dim3(NROW / 16, MCHUNK_BLKS), 32, 0, stream>>>(adjs, xw2, agg, mchunk);
    k_logsoftmax<<<(NROW + 255) / 256, 256, 0, stream>>>(agg, b2, out);
}